// HMMNeuronLayer_14413910245586
// MI455X (gfx1250) — compile-verified
//
#include <hip/hip_runtime.h>
#include <hip/hip_bf16.h>
#include <math.h>

typedef __attribute__((ext_vector_type(2))) float v2f;
typedef __attribute__((ext_vector_type(8))) float v8f;

#define BATCH  512
#define TLEN   4096
#define NSTATE 16
#define HALF_L2PI 0.9189385332046727f

// workspace layout (bytes); total ~17.8 MB
#define OFF_LOGA   0u          // 256 f32      : normalized log transition matrix
#define OFF_LOGPI  1024u       // 16 f32       : log initial distribution
#define OFF_DELTA  2048u       // 512*16 f32   : final deltas per sequence
#define OFF_G      36864u      // 512*64*16 B  : per-chunk backtrack maps
#define OFF_S      561152u     // 512*64  B    : chunk-boundary states
#define OFF_PSI    1048576u    // 512*2048*16 B: psi, 2 timesteps packed per byte

__device__ __forceinline__ float bperm(float v, int byteIdx) {
  return __int_as_float(__builtin_amdgcn_ds_bpermute(byteIdx, __float_as_int(v)));
}

// ---------------------------------------------------------------------------
// Preprocess: log_A and log_pi. Row sums of trans via WMMA (trans x ones).
// One wave; EXEC all ones before the WMMA ops.
// ---------------------------------------------------------------------------
__global__ __launch_bounds__(32) void k_pre(const float* __restrict__ hp,
                                            void* __restrict__ wsv) {
  char* ws = (char*)wsv;
  float* logA  = (float*)(ws + OFF_LOGA);
  float* logpi = (float*)(ws + OFF_LOGPI);

  const int lane = threadIdx.x;      // 0..31
  const int half = lane >> 4;
  const int r    = lane & 15;

  v8f acc = {};
  v2f onesB = {1.0f, 1.0f};
#pragma unroll
  for (int kc = 0; kc < 4; ++kc) {
    // A-frag (16x4 f32): lanes 0-15 hold K=0,1 ; lanes 16-31 hold K=2,3
    const int kb = kc * 4 + 2 * half;
    v2f A;
    A.x = hp[r * 16 + kb];
    A.y = hp[r * 16 + kb + 1];
    acc = __builtin_amdgcn_wmma_f32_16x16x4_f32(
        /*neg_a=*/false, A, /*neg_b=*/false, onesB,
        /*c_mod=*/(short)0, acc, /*reuse_a=*/false, /*reuse_b=*/false);
  }
  // D layout: acc[v] = rowsum[v + 8*half] (same in every column N=r)
#pragma unroll
  for (int v = 0; v < 8; ++v) {
    const int i = v + 8 * half;
    logA[i * 16 + r] = logf(hp[i * 16 + r]) - logf(acc[v]);
  }
  if (half == 0) {
    // init = trans[0,:]; its sum is rowsum[0] = acc[0] on lanes 0-15
    logpi[r] = logf(hp[r]) - logf(acc[0]);
  }
}

// ---------------------------------------------------------------------------
// Forward Viterbi. One wave (32 threads) per block handles 4 sequences:
// chain0 -> seqs base+0 (lanes 0-15) / base+1 (lanes 16-31)
// chain1 -> seqs base+2 / base+3 (interleaved for ILP).
// Lane owns state j = lane&15; delta[i] broadcast via ds_bpermute within half.
// psi packed 4-bit, 2 timesteps per byte.
// ---------------------------------------------------------------------------
__global__ __launch_bounds__(32) void k_fwd(const float* __restrict__ in,
                                            void* __restrict__ wsv) {
  char* ws = (char*)wsv;
  const float* __restrict__ logA  = (const float*)(ws + OFF_LOGA);
  const float* __restrict__ logpi = (const float*)(ws + OFF_LOGPI);
  float* __restrict__ dfin        = (float*)(ws + OFF_DELTA);
  unsigned char* __restrict__ psi = (unsigned char*)(ws + OFF_PSI);

  const int lane = threadIdx.x;
  const int j = lane & 15;
  const int permBase = (lane & 16) * 4;        // byte index base for half-wave bcast
  const int s0 = blockIdx.x * 4 + (lane >> 4); // chain 0 sequence
  const int s1 = s0 + 2;                       // chain 1 sequence

  float a[16];
#pragma unroll
  for (int i = 0; i < 16; ++i) a[i] = logA[i * 16 + j];  // column j of log_A
  const float lp = logpi[j];

  const float* __restrict__ xp0 = in + (size_t)s0 * TLEN;
  const float* __restrict__ xp1 = in + (size_t)s1 * TLEN;
  unsigned char* __restrict__ psi0 = psi + (size_t)s0 * (2048 * 16) + j;
  unsigned char* __restrict__ psi1 = psi + (size_t)s1 * (2048 * 16) + j;

  float d0, d1;
  unsigned int low0 = 0, low1 = 0;
  int pb0 = 0, pb1 = 0;

  auto core = [&](float e0, float e1) {
    float m0 = -INFINITY, m1 = -INFINITY;
    int b0 = 0, b1 = 0;
#pragma unroll
    for (int i = 0; i < 16; ++i) {
      const int bidx = permBase + i * 4;
      float di0 = bperm(d0, bidx);
      float di1 = bperm(d1, bidx);
      float sc0 = di0 + a[i];
      float sc1 = di1 + a[i];
      bool g0 = sc0 > m0;                 // strict > : first occurrence on ties
      bool g1 = sc1 > m1;
      m0 = g0 ? sc0 : m0;  b0 = g0 ? i : b0;
      m1 = g1 ? sc1 : m1;  b1 = g1 ? i : b1;
    }
    d0 = m0 + e0;
    d1 = m1 + e1;
    pb0 = b0; pb1 = b1;
  };

  { // block t0 = 0 : init at t=0, transitions t=1..15
    float xv0 = xp0[j], xv1 = xp1[j];
    float ev0 = -0.5f * (xv0 * xv0) - HALF_L2PI;
    float ev1 = -0.5f * (xv1 * xv1) - HALF_L2PI;
    d0 = lp + bperm(ev0, permBase);      // emit at t=0
    d1 = lp + bperm(ev1, permBase);
    // t = 1 (odd)
    core(bperm(ev0, permBase + 4), bperm(ev1, permBase + 4));
    psi0[0] = (unsigned char)(low0 | (pb0 << 4));
    psi1[0] = (unsigned char)(low1 | (pb1 << 4));
#pragma unroll 2
    for (int k = 2; k < 16; k += 2) {
      core(bperm(ev0, permBase + k * 4), bperm(ev1, permBase + k * 4));
      low0 = (unsigned)pb0; low1 = (unsigned)pb1;
      core(bperm(ev0, permBase + (k + 1) * 4), bperm(ev1, permBase + (k + 1) * 4));
      psi0[(k >> 1) * 16] = (unsigned char)(low0 | (pb0 << 4));
      psi1[(k >> 1) * 16] = (unsigned char)(low1 | (pb1 << 4));
    }
  }

  for (int t0 = 16; t0 < TLEN; t0 += 16) {
    __builtin_prefetch(xp0 + t0 + 16, 0, 1);
    __builtin_prefetch(xp1 + t0 + 16, 0, 1);
    float xv0 = xp0[t0 + j], xv1 = xp1[t0 + j];
    float ev0 = -0.5f * (xv0 * xv0) - HALF_L2PI;
    float ev1 = -0.5f * (xv1 * xv1) - HALF_L2PI;
    const int pairBase = t0 >> 1;
#pragma unroll 2
    for (int k = 0; k < 16; k += 2) {
      core(bperm(ev0, permBase + k * 4), bperm(ev1, permBase + k * 4));          // even t
      low0 = (unsigned)pb0; low1 = (unsigned)pb1;
      core(bperm(ev0, permBase + (k + 1) * 4), bperm(ev1, permBase + (k + 1) * 4)); // odd t
      const int pi = pairBase + (k >> 1);
      psi0[pi * 16] = (unsigned char)(low0 | (pb0 << 4));
      psi1[pi * 16] = (unsigned char)(low1 | (pb1 << 4));
    }
  }

  dfin[s0 * 16 + j] = d0;
  dfin[s1 * 16 + j] = d1;
}

// ---------------------------------------------------------------------------
// Backtrack phase 1: per (b, chunk c=1..63), 16 lanes each walk 64 steps from
// every start state, producing map G_c: z at t=(c+1)*64-1  ->  z at t=c*64-1.
// ---------------------------------------------------------------------------
__global__ __launch_bounds__(256) void k_back1(void* __restrict__ wsv) {
  char* ws = (char*)wsv;
  const unsigned char* __restrict__ psi = (const unsigned char*)(ws + OFF_PSI);
  unsigned char* __restrict__ G = (unsigned char*)(ws + OFF_G);

  const int tid = blockIdx.x * blockDim.x + threadIdx.x;
  const int jj  = tid & 15;
  const int grp = tid >> 4;            // 0 .. 512*63-1
  const int b = grp / 63;
  const int c = grp % 63 + 1;          // chunks 1..63
  if (b >= BATCH) return;

  const unsigned char* __restrict__ pb = psi + (size_t)b * (2048 * 16);
  int z = jj;
  const int thi = (c + 1) * 64 - 1;
  for (int t = thi; t >= c * 64; --t) {
    unsigned char byte = pb[(t >> 1) * 16 + z];
    z = (byte >> ((t & 1) * 4)) & 15;
  }
  G[((size_t)b * 64 + c) * 16 + jj] = (unsigned char)z;
}

// ---------------------------------------------------------------------------
// Backtrack phase 2: per batch b, argmax(delta_T) then compose the 63 chunk
// maps sequentially, recording the boundary state S_c = z at t=(c+1)*64-1.
// ---------------------------------------------------------------------------
__global__ __launch_bounds__(256) void k_back2(void* __restrict__ wsv) {
  char* ws = (char*)wsv;
  const float* __restrict__ dfin = (const float*)(ws + OFF_DELTA);
  const unsigned char* __restrict__ G = (const unsigned char*)(ws + OFF_G);
  unsigned char* __restrict__ Sb = (unsigned char*)(ws + OFF_S);

  const int b = blockIdx.x * blockDim.x + threadIdx.x;
  if (b >= BATCH) return;

  float m = dfin[b * 16];
  int z = 0;
#pragma unroll
  for (int jj = 1; jj < 16; ++jj) {    // first occurrence (strict >)
    float v = dfin[b * 16 + jj];
    if (v > m) { m = v; z = jj; }
  }
  for (int c = 63; c >= 1; --c) {
    Sb[b * 64 + c] = (unsigned char)z;
    z = G[((size_t)b * 64 + c) * 16 + z];
  }
  Sb[b * 64 + 0] = (unsigned char)z;
}

// ---------------------------------------------------------------------------
// Backtrack phase 3: per (b, chunk), re-walk 64 steps from the known boundary
// state and emit the path. 32768 independent threads -> latency fully hidden.
// ---------------------------------------------------------------------------
__global__ __launch_bounds__(256) void k_back3(void* __restrict__ wsv,
                                               int* __restrict__ out) {
  char* ws = (char*)wsv;
  const unsigned char* __restrict__ psi = (const unsigned char*)(ws + OFF_PSI);
  const unsigned char* __restrict__ Sb  = (const unsigned char*)(ws + OFF_S);

  const int tid = blockIdx.x * blockDim.x + threadIdx.x;
  const int b = tid >> 6;
  const int c = tid & 63;
  if (b >= BATCH) return;

  const unsigned char* __restrict__ pb = psi + (size_t)b * (2048 * 16);
  int* __restrict__ ob = out + (size_t)b * TLEN;

  int z = Sb[b * 64 + c];
  const int thi = (c + 1) * 64 - 1;
  ob[thi] = z;
  const int tlo = (c == 0) ? 1 : c * 64 + 1;
  for (int t = thi; t >= tlo; --t) {
    unsigned char byte = pb[(t >> 1) * 16 + z];
    z = (byte >> ((t & 1) * 4)) & 15;
    ob[t - 1] = z;
  }
}

// ---------------------------------------------------------------------------
extern "C" void kernel_launch(void* const* d_in, const int* in_sizes, int n_in,
                              void* d_out, int out_size, void* d_ws, size_t ws_size,
                              hipStream_t stream) {
  (void)in_sizes; (void)n_in; (void)out_size; (void)ws_size;
  const float* inputs = (const float*)d_in[0];   // (512, 4096) f32
  const float* hp     = (const float*)d_in[1];   // (64, 16, 16) f32 (only [0] used)
  int* out = (int*)d_out;                        // (512, 4096) int32

  k_pre  <<<1,                 32, 0, stream>>>(hp, d_ws);
  k_fwd  <<<BATCH / 4,         32, 0, stream>>>(inputs, d_ws);
  k_back1<<<(BATCH * 63 * 16) / 256, 256, 0, stream>>>(d_ws);
  k_back2<<<(BATCH + 255) / 256,     256, 0, stream>>>(d_ws);
  k_back3<<<(BATCH * 64) / 256,      256, 0, stream>>>(d_ws, out);
}